// SparseEmbedding_19310172962874
// MI455X (gfx1250) — compile-verified
//
#include <hip/hip_runtime.h>
#include <stdint.h>

// Embedding gather: out[row, :] = weight[indices[row], :]
//   indices : int32 [BATCH*HIST]        (819,200 rows)
//   weight  : fp32  [1,000,000 x 128]   (512 MB table, random access)
//   out     : fp32  [BATCH*HIST x 128]  (419 MB streaming writes)
//
// Pure bandwidth kernel: ~0.85 GB total traffic -> ~36 us at 23.3 TB/s.
// Zero FLOPs, so no WMMA. CDNA5-specific paths used instead:
//   * global_load_async_to_lds_b32 + s_wait_asynccnt (async index staging)
//   * b128 vector loads, non-temporal b128 stores (protect L2 for weight reuse)

typedef float v4f __attribute__((ext_vector_type(4)));   // native vector: NT-store legal

#define WAVES_PER_BLOCK 8
#define ROWS_PER_WAVE   4
#define ROWS_PER_BLOCK  (WAVES_PER_BLOCK * ROWS_PER_WAVE)   // 32
#define F4_PER_ROW      32                                   // 128 floats = 32 x v4f

__global__ __launch_bounds__(256) void SparseEmbedding_gather_kernel(
    const int* __restrict__ indices,
    const float* __restrict__ weight,
    float* __restrict__ out,
    int num_rows)
{
    __shared__ int s_idx[ROWS_PER_BLOCK];

    const int tid       = threadIdx.x;
    const int blockRow0 = blockIdx.x * ROWS_PER_BLOCK;

    // ---- Stage this block's 32 row indices into LDS via the async data mover.
    // Wave 0's 32 lanes each issue one async memory->LDS transfer (no VGPR
    // writeback), then drain ASYNCcnt before the block barrier.
    if (tid < ROWS_PER_BLOCK) {
        const int row = blockRow0 + tid;
        if (row < num_rows) {
            // LDS flat address truncates to the LDS byte offset (ISA 10.2,
            // LDS aperture: low 32 bits of the address).
            uint32_t   lds_off = (uint32_t)(uintptr_t)(&s_idx[tid]);
            const int* gsrc    = indices + row;
            asm volatile("global_load_async_to_lds_b32 %0, %1, off"
                         :
                         : "v"(lds_off), "v"(gsrc)
                         : "memory");
            asm volatile("s_wait_asynccnt 0" ::: "memory");
        } else {
            s_idx[tid] = 0;
        }
    }
    __syncthreads();

    const int wave = tid >> 5;   // 0..7
    const int lane = tid & 31;   // one 16B slice of the 512B row per lane

    const v4f* __restrict__ w4 = reinterpret_cast<const v4f*>(weight);
    v4f*       __restrict__ o4 = reinterpret_cast<v4f*>(out);

    const int rowBase = blockRow0 + wave * ROWS_PER_WAVE;

    if (rowBase + ROWS_PER_WAVE <= num_rows) {
        // Fast path: gather indices (wave-uniform ds_loads), then issue all 4
        // independent b128 loads before any store so LOADcnt pipelines the
        // random-access latency of the 512MB table.
        int idx[ROWS_PER_WAVE];
#pragma unroll
        for (int r = 0; r < ROWS_PER_WAVE; ++r)
            idx[r] = s_idx[wave * ROWS_PER_WAVE + r];

        v4f v[ROWS_PER_WAVE];
#pragma unroll
        for (int r = 0; r < ROWS_PER_WAVE; ++r)
            v[r] = w4[(size_t)idx[r] * F4_PER_ROW + lane];

        // Streaming output: non-temporal so 419MB of writes do not evict the
        // weight rows that get L2 reuse (~30% expected hit rate).
#pragma unroll
        for (int r = 0; r < ROWS_PER_WAVE; ++r)
            __builtin_nontemporal_store(
                v[r], &o4[(size_t)(rowBase + r) * F4_PER_ROW + lane]);
    } else {
        // Tail blocks (not hit for 819,200 rows, kept for generality).
        for (int r = 0; r < ROWS_PER_WAVE; ++r) {
            const int row = rowBase + r;
            if (row < num_rows) {
                const int idx = s_idx[wave * ROWS_PER_WAVE + r];
                v4f v = w4[(size_t)idx * F4_PER_ROW + lane];
                __builtin_nontemporal_store(
                    v, &o4[(size_t)row * F4_PER_ROW + lane]);
            }
        }
    }
}

extern "C" void kernel_launch(void* const* d_in, const int* in_sizes, int n_in,
                              void* d_out, int out_size, void* d_ws, size_t ws_size,
                              hipStream_t stream) {
    const int*   indices = (const int*)d_in[0];    // [BATCH*HIST] int32
    const float* weight  = (const float*)d_in[1];  // [NUM_EMB, 128] fp32
    float*       out     = (float*)d_out;          // [BATCH*HIST, 128] fp32

    const int num_rows = in_sizes[0];              // 4096 * 200 = 819,200
    const int blocks   = (num_rows + ROWS_PER_BLOCK - 1) / ROWS_PER_BLOCK;

    SparseEmbedding_gather_kernel<<<blocks, 256, 0, stream>>>(
        indices, weight, out, num_rows);
}